// HyperConcatSquash_40312563040401
// MI455X (gfx1250) — compile-verified
//
#include <hip/hip_runtime.h>
#include <math.h>

// ---------------------------------------------------------------------------
// HyperConcatSquash for MI455X (gfx1250, wave32, WMMA).
//
// Pipeline (all f32, all WMMA f32 16x16x4):
//   K1..K3: hyper_net  MLP  (c: 64x128) -> W   (64 x 64 x 64) in ws
//   K4..K6: hyper_gate MLP  (t: 64x32)  -> a   (64 x 64)      in ws (sigmoid)
//   K7..K9: hyper_bias MLP  (t: 64x32)  -> b   (64 x 64)      in ws
//   K10:    ragged matmul   y[n] = a[s]*(x[n] @ W[s]^T) + b[s], s = batch[n]
//
// batch = repeat(arange(64), 512): sorted, 512 rows per segment, so every
// 128-row block belongs to exactly one segment (read from batch[] anyway).
// ---------------------------------------------------------------------------

typedef float v2f __attribute__((ext_vector_type(2)));
typedef float v8f __attribute__((ext_vector_type(8)));

#define ACT_NONE    0
#define ACT_BN_RELU 1
#define ACT_SIGMOID 2

#define BN_EPS 1e-5f

static __device__ inline v8f v8f_zero() {
  v8f z = {0.f, 0.f, 0.f, 0.f, 0.f, 0.f, 0.f, 0.f};
  return z;
}

// D = A(16x4) * B(4x16) + C, f32 tensor op.
static __device__ inline v8f wmma_f32_4(v2f a, v2f b, v8f c) {
  return __builtin_amdgcn_wmma_f32_16x16x4_f32(
      /*neg_a=*/false, a, /*neg_b=*/false, b,
      /*c_mod=*/(short)0, c, /*reuse_a=*/false, /*reuse_b=*/false);
}

// ---------------------------------------------------------------------------
// Fused hypernet layer: Y[64 x N] = act( X[64 x K] @ Wt[N x K]^T + bias[N] )
// One block = 64 output columns x all 64 batch rows (so BatchNorm stats are
// computed entirely in-block). 128 threads = 4 waves; wave w owns M-tile w.
// K-loop is unrolled x2 with ALL fragment loads (10 global_load_b64) issued
// before the 8 WMMAs so the waits drain incrementally instead of stalling
// loadcnt==0 before every WMMA.
// act: 0 = none, 1 = train-mode BatchNorm + ReLU, 2 = sigmoid.
// Requires K % 8 == 0 (true for K = 32, 128, 512).
// ---------------------------------------------------------------------------
__global__ __launch_bounds__(128) void hyper_gemm_kernel(
    const float* __restrict__ X, const float* __restrict__ Wt,
    const float* __restrict__ bias, const float* __restrict__ gamma,
    const float* __restrict__ beta, float* __restrict__ Y,
    int K, int N, int act) {
  const int tid   = threadIdx.x;
  const int lane  = tid & 31;
  const int wave  = tid >> 5;          // M-tile index (0..3)
  const int ncol0 = blockIdx.x * 64;   // first output column of this block
  const int lrow  = lane & 15;         // row/col within 16-wide tile
  const int khalf = (lane >> 4) << 1;  // K offset contributed by lane half

  __shared__ float sZ[64][65];         // pre-activation staging (padded)
  __shared__ float sScale[64];
  __shared__ float sShift[64];

  v8f acc[4];
#pragma unroll
  for (int nt = 0; nt < 4; ++nt) acc[nt] = v8f_zero();

  const int m = wave * 16 + lrow;      // A-fragment row for this lane
  for (int k = 0; k < K; k += 8) {
    const int kb0 = k + khalf;
    const int kb1 = k + 4 + khalf;

    // Batch-issue all loads for two K-steps before any WMMA.
    v2f af0, af1;
    af0.x = X[m * K + kb0];
    af0.y = X[m * K + kb0 + 1];
    af1.x = X[m * K + kb1];
    af1.y = X[m * K + kb1 + 1];

    v2f bf0[4], bf1[4];
#pragma unroll
    for (int nt = 0; nt < 4; ++nt) {
      const int n = ncol0 + nt * 16 + lrow;
      bf0[nt].x = Wt[n * K + kb0];
      bf0[nt].y = Wt[n * K + kb0 + 1];
      bf1[nt].x = Wt[n * K + kb1];
      bf1[nt].y = Wt[n * K + kb1 + 1];
    }

#pragma unroll
    for (int nt = 0; nt < 4; ++nt) acc[nt] = wmma_f32_4(af0, bf0[nt], acc[nt]);
#pragma unroll
    for (int nt = 0; nt < 4; ++nt) acc[nt] = wmma_f32_4(af1, bf1[nt], acc[nt]);
  }

  // Spill accumulators (+bias) to LDS. C/D layout: lanes 0-15 -> M = r,
  // lanes 16-31 -> M = r + 8 (within the wave's 16-row tile).
  const int mofs = (lane >> 4) << 3;
#pragma unroll
  for (int nt = 0; nt < 4; ++nt) {
#pragma unroll
    for (int r = 0; r < 8; ++r) {
      const int mm = wave * 16 + r + mofs;
      const int nn = nt * 16 + lrow;
      sZ[mm][nn] = acc[nt][r] + bias[ncol0 + nn];
    }
  }
  __syncthreads();

  if (act == ACT_BN_RELU) {
    if (tid < 64) {
      float s = 0.f, s2 = 0.f;
#pragma unroll 8
      for (int mm = 0; mm < 64; ++mm) {
        const float v = sZ[mm][tid];
        s += v;
        s2 += v * v;
      }
      const float mean = s * (1.f / 64.f);
      const float var  = s2 * (1.f / 64.f) - mean * mean;
      const float sc   = gamma[ncol0 + tid] * rsqrtf(var + BN_EPS);
      sScale[tid] = sc;
      sShift[tid] = beta[ncol0 + tid] - mean * sc;
    }
    __syncthreads();
  }

  for (int i = tid; i < 64 * 64; i += 128) {
    const int mm = i >> 6;
    const int nn = i & 63;
    float v = sZ[mm][nn];
    if (act == ACT_BN_RELU) {
      v = v * sScale[nn] + sShift[nn];
      v = v > 0.f ? v : 0.f;
    } else if (act == ACT_SIGMOID) {
      v = 1.f / (1.f + __expf(-v));
    }
    Y[mm * N + ncol0 + nn] = v;
  }
}

// ---------------------------------------------------------------------------
// Ragged per-segment matmul: out[n,:] = a[s] * (x[n,:] @ W[s]^T) + b[s].
// 256 blocks x 128 rows; each block stages W_s and its x-tile in padded LDS
// and runs 16 K-steps of 8 WMMAs per wave (wave w owns M-tiles 2w, 2w+1).
// All 6 LDS fragment loads are issued before the 8-WMMA burst.
// ---------------------------------------------------------------------------
__global__ __launch_bounds__(128) void ragged_matmul_kernel(
    const float* __restrict__ x, const int* __restrict__ batch,
    const float* __restrict__ Wbuf, const float* __restrict__ abuf,
    const float* __restrict__ bbuf, float* __restrict__ out) {
  const int tid   = threadIdx.x;
  const int lane  = tid & 31;
  const int wave  = tid >> 5;
  const int row0  = blockIdx.x * 128;
  const int lrow  = lane & 15;
  const int khalf = (lane >> 4) << 1;

  __shared__ float sX[128][68];  // x tile, padded to kill bank conflicts
  __shared__ float sW[64][68];   // W_s (row-major [out][in]), padded
  __shared__ float sA[64];
  __shared__ float sB[64];

  const int seg = batch[row0];

  for (int i = tid; i < 128 * 64; i += 128) {
    sX[i >> 6][i & 63] = x[(size_t)row0 * 64 + i];
  }
  for (int i = tid; i < 64 * 64; i += 128) {
    sW[i >> 6][i & 63] = Wbuf[seg * 4096 + i];
  }
  if (tid < 64) {
    sA[tid] = abuf[seg * 64 + tid];
    sB[tid] = bbuf[seg * 64 + tid];
  }
  __syncthreads();

  v8f acc[2][4];
#pragma unroll
  for (int mt = 0; mt < 2; ++mt)
#pragma unroll
    for (int nt = 0; nt < 4; ++nt) acc[mt][nt] = v8f_zero();

  for (int k = 0; k < 64; k += 4) {
    const int kb = k + khalf;
    v2f af[2];
#pragma unroll
    for (int mt = 0; mt < 2; ++mt) {
      const int ml = wave * 32 + mt * 16 + lrow;
      af[mt].x = sX[ml][kb];
      af[mt].y = sX[ml][kb + 1];
    }
    v2f bf[4];
#pragma unroll
    for (int nt = 0; nt < 4; ++nt) {
      const int n = nt * 16 + lrow;
      bf[nt].x = sW[n][kb];
      bf[nt].y = sW[n][kb + 1];
    }
#pragma unroll
    for (int mt = 0; mt < 2; ++mt)
#pragma unroll
      for (int nt = 0; nt < 4; ++nt)
        acc[mt][nt] = wmma_f32_4(af[mt], bf[nt], acc[mt][nt]);
  }

  const int mofs = (lane >> 4) << 3;
#pragma unroll
  for (int mt = 0; mt < 2; ++mt) {
#pragma unroll
    for (int nt = 0; nt < 4; ++nt) {
#pragma unroll
      for (int r = 0; r < 8; ++r) {
        const int ml = wave * 32 + mt * 16 + r + mofs;
        const int nn = nt * 16 + lrow;
        const float v = sA[nn] * acc[mt][nt][r] + sB[nn];
        out[(size_t)(row0 + ml) * 64 + nn] = v;
      }
    }
  }
}

// ---------------------------------------------------------------------------
// Host-side launch. Input order = setup_inputs() dict flattened JAX-style
// (sorted dict keys):
//   0: x[32768*64]  1: c[64*128]  2: t[64*32]  3: batch[32768]
//   params -> hyper_bias, hyper_gate, hyper_net; within each MLP the sorted
//   key order is W1, W2, W3, b1, b2, b3, be1, be2, g1, g2.
// Workspace (f32): Wbuf 262144 | abuf 4096 | bbuf 4096 | H1 32768 | H2 32768
// (~1.32 MB total).
// ---------------------------------------------------------------------------
extern "C" void kernel_launch(void* const* d_in, const int* in_sizes, int n_in,
                              void* d_out, int out_size, void* d_ws,
                              size_t ws_size, hipStream_t stream) {
  const float* x     = (const float*)d_in[0];  // 2097152
  const float* c     = (const float*)d_in[1];  // 8192
  const float* t     = (const float*)d_in[2];  // 2048
  const int*   batch = (const int*)d_in[3];    // 32768

  // hyper_bias (T_DIM=32 -> 64): W1[512*32] W2[512*512] W3[64*512] ...
  const float* hb_W1  = (const float*)d_in[4];
  const float* hb_W2  = (const float*)d_in[5];
  const float* hb_W3  = (const float*)d_in[6];
  const float* hb_b1  = (const float*)d_in[7];
  const float* hb_b2  = (const float*)d_in[8];
  const float* hb_b3  = (const float*)d_in[9];
  const float* hb_be1 = (const float*)d_in[10];
  const float* hb_be2 = (const float*)d_in[11];
  const float* hb_g1  = (const float*)d_in[12];
  const float* hb_g2  = (const float*)d_in[13];

  // hyper_gate (32 -> 64)
  const float* hg_W1  = (const float*)d_in[14];
  const float* hg_W2  = (const float*)d_in[15];
  const float* hg_W3  = (const float*)d_in[16];
  const float* hg_b1  = (const float*)d_in[17];
  const float* hg_b2  = (const float*)d_in[18];
  const float* hg_b3  = (const float*)d_in[19];
  const float* hg_be1 = (const float*)d_in[20];
  const float* hg_be2 = (const float*)d_in[21];
  const float* hg_g1  = (const float*)d_in[22];
  const float* hg_g2  = (const float*)d_in[23];

  // hyper_net (CTX_DIM=128 -> 4096): W1[512*128] W2[512*512] W3[4096*512]
  const float* hn_W1  = (const float*)d_in[24];
  const float* hn_W2  = (const float*)d_in[25];
  const float* hn_W3  = (const float*)d_in[26];
  const float* hn_b1  = (const float*)d_in[27];
  const float* hn_b2  = (const float*)d_in[28];
  const float* hn_b3  = (const float*)d_in[29];
  const float* hn_be1 = (const float*)d_in[30];
  const float* hn_be2 = (const float*)d_in[31];
  const float* hn_g1  = (const float*)d_in[32];
  const float* hn_g2  = (const float*)d_in[33];

  float* ws   = (float*)d_ws;
  float* Wbuf = ws;                  // 64*64*64
  float* abuf = Wbuf + 262144;       // 64*64
  float* bbuf = abuf + 4096;         // 64*64
  float* H1   = bbuf + 4096;         // 64*512
  float* H2   = H1 + 32768;          // 64*512

  const dim3 blk(128);

  // hyper_net: c -> W
  hyper_gemm_kernel<<<8, blk, 0, stream>>>(c, hn_W1, hn_b1, hn_g1, hn_be1, H1,
                                           128, 512, ACT_BN_RELU);
  hyper_gemm_kernel<<<8, blk, 0, stream>>>(H1, hn_W2, hn_b2, hn_g2, hn_be2, H2,
                                           512, 512, ACT_BN_RELU);
  hyper_gemm_kernel<<<64, blk, 0, stream>>>(H2, hn_W3, hn_b3, nullptr, nullptr,
                                            Wbuf, 512, 4096, ACT_NONE);

  // hyper_gate: t -> a (sigmoid)
  hyper_gemm_kernel<<<8, blk, 0, stream>>>(t, hg_W1, hg_b1, hg_g1, hg_be1, H1,
                                           32, 512, ACT_BN_RELU);
  hyper_gemm_kernel<<<8, blk, 0, stream>>>(H1, hg_W2, hg_b2, hg_g2, hg_be2, H2,
                                           512, 512, ACT_BN_RELU);
  hyper_gemm_kernel<<<1, blk, 0, stream>>>(H2, hg_W3, hg_b3, nullptr, nullptr,
                                           abuf, 512, 64, ACT_SIGMOID);

  // hyper_bias: t -> b
  hyper_gemm_kernel<<<8, blk, 0, stream>>>(t, hb_W1, hb_b1, hb_g1, hb_be1, H1,
                                           32, 512, ACT_BN_RELU);
  hyper_gemm_kernel<<<8, blk, 0, stream>>>(H1, hb_W2, hb_b2, hb_g2, hb_be2, H2,
                                           512, 512, ACT_BN_RELU);
  hyper_gemm_kernel<<<1, blk, 0, stream>>>(H2, hb_W3, hb_b3, nullptr, nullptr,
                                           bbuf, 512, 64, ACT_NONE);

  // Main ragged matmul + gate/bias epilogue.
  ragged_matmul_kernel<<<256, blk, 0, stream>>>(x, batch, Wbuf, abuf, bbuf,
                                                (float*)d_out);
}